// SeasonWeightedLoss_55963423866919
// MI455X (gfx1250) — compile-verified
//
#include <hip/hip_runtime.h>
#include <stdint.h>

// ---------------- problem constants ----------------
#define TPB        256
#define NBLK       2048
#define NROWS      2097152u          // B = 2^21, == NBLK*TPB*4 exactly
#define NCLS       16
#define ROWS_PER_T 4
#define LDS_ROW_F  20                // 16 data floats + 4 pad -> 80B row, bank-conflict-free

typedef __attribute__((ext_vector_type(2))) float v2f;
typedef __attribute__((ext_vector_type(8))) float v8f;

// ---- CDNA5 async global->LDS copy of one 64B row (4 x b128), per-lane ----
// VDST vgpr = LDS byte offset (low 32 bits of the generic shared pointer),
// VADDR = 64-bit global address, GV mode (saddr = off).
__device__ __forceinline__ void async_row_copy(const float* gsrc, void* ldst) {
  uint32_t loff = (uint32_t)(uintptr_t)ldst;   // generic LDS ptr: low 32 bits = LDS offset
#pragma unroll
  for (int q = 0; q < 4; ++q) {
    uint32_t lo = loff + (uint32_t)(q * 16);
    const float* g = gsrc + q * 4;
    asm volatile("global_load_async_to_lds_b128 %0, %1, off"
                 :: "v"(lo), "v"(g)
                 : "memory");
  }
}

__device__ __forceinline__ void wait_async_le4() {
  asm volatile("s_wait_asynccnt 4" ::: "memory");
}
__device__ __forceinline__ void wait_async_0() {
  asm volatile("s_wait_asynccnt 0" ::: "memory");
}

// ================= main streaming kernel =================
// Each thread owns ROWS_PER_T rows (grid-stride), stages its row through a
// double-buffered LDS tile via async copies, computes weighted NLL, then
// wave32 + block reduction -> one partial per block in d_ws.
__global__ __launch_bounds__(TPB) void season_loss_main(
    const float* __restrict__ x, const int* __restrict__ tgt,
    const float* __restrict__ wmat, float* __restrict__ partial)
{
  __shared__ float wlds[NCLS * NCLS];                       // 1 KB weight table
  __shared__ __align__(16) float tile[2][TPB * LDS_ROW_F];  // 2 x 20 KB staging

  const int tid = threadIdx.x;
  wlds[tid] = wmat[tid];                                    // TPB == 256 == NCLS*NCLS

  const size_t stride = (size_t)NBLK * TPB;                 // 524288 rows
  size_t r = (size_t)blockIdx.x * TPB + tid;

  float* buf0 = &tile[0][tid * LDS_ROW_F];
  float* buf1 = &tile[1][tid * LDS_ROW_F];

  async_row_copy(x + r * NCLS, buf0);                       // prime buffer 0
  __syncthreads();                                          // wlds visible

  float acc = 0.0f;
#pragma unroll
  for (int it = 0; it < ROWS_PER_T; ++it) {
    const size_t rn = r + stride;
    if (it < ROWS_PER_T - 1) {
      async_row_copy(x + rn * NCLS, (it & 1) ? buf0 : buf1);  // prefetch next tile
      wait_async_le4();                                       // retire current tile's 4 copies
    } else {
      wait_async_0();
    }
    const float* xr = (it & 1) ? buf1 : buf0;

    // pull the row into registers (conflict-free ds_load_b128 x4: 80B row stride)
    float4 q0 = *(const float4*)(xr + 0);
    float4 q1 = *(const float4*)(xr + 4);
    float4 q2 = *(const float4*)(xr + 8);
    float4 q3 = *(const float4*)(xr + 12);
    float v[NCLS] = {q0.x, q0.y, q0.z, q0.w,  q1.x, q1.y, q1.z, q1.w,
                     q2.x, q2.y, q2.z, q2.w,  q3.x, q3.y, q3.z, q3.w};

    // max + first-occurrence argmax (strict > keeps lowest index, matches jnp.argmax)
    float m = v[0]; int am = 0;
#pragma unroll
    for (int j = 1; j < NCLS; ++j) { if (v[j] > m) { m = v[j]; am = j; } }

    float s = 0.0f;
#pragma unroll
    for (int j = 0; j < NCLS; ++j) s += __expf(v[j] - m);

    const int t = tgt[r];
    const float xt = xr[t];                                 // dynamic index -> 1 ds_load_b32
    const float loss = m + __logf(s) - xt;                  // = logsumexp - x[t]
    const float w = (am == t) ? 1.0f : wlds[t * NCLS + am];
    acc += loss * w;

    r = rn;
  }

  // wave32 reduction
#pragma unroll
  for (int off = 16; off > 0; off >>= 1) acc += __shfl_down(acc, off, 32);

  __shared__ float wsum[TPB / 32];
  if ((tid & 31) == 0) wsum[tid >> 5] = acc;
  __syncthreads();
  if (tid == 0) {
    float sblk = 0.0f;
#pragma unroll
    for (int i = 0; i < TPB / 32; ++i) sblk += wsum[i];
    partial[blockIdx.x] = sblk;
  }
}

// ================= final reduction via V_WMMA_F32_16X16X4_F32 =================
// One wave32, EXEC all ones. Treat each chunk of 64 partials as A(16x4) f32;
// B = ones(4x16) => D[i][j] = rowsum_i (exact f32 adds), accumulated in C over
// 32 chunks. Column 0 of C: rows 0-7 live in lane 0's 8 VGPRs, rows 8-15 in
// lane 16's. Deterministic, exact.
__global__ __launch_bounds__(32) void season_loss_final(
    const float* __restrict__ partial, float* __restrict__ out)
{
  const int lane = threadIdx.x;
  // A-matrix f32 16x4 layout: lanes 0-15 hold (M=lane, K=0/1) in vgpr0/1,
  // lanes 16-31 hold (M=lane-16, K=2/3). A[M][K] = p[M*4+K].
  const int base = (lane < 16) ? (lane * 4) : ((lane - 16) * 4 + 2);

  v2f b; b.x = 1.0f; b.y = 1.0f;   // ones: layout-independent
  v8f c = {0.f, 0.f, 0.f, 0.f, 0.f, 0.f, 0.f, 0.f};

  for (int chunk = 0; chunk < NBLK; chunk += 64) {
    v2f a;
    a.x = partial[chunk + base];
    a.y = partial[chunk + base + 1];
    c = __builtin_amdgcn_wmma_f32_16x16x4_f32(
        /*neg_a=*/false, a, /*neg_b=*/false, b,
        /*c_mod=*/(short)0, c, /*reuse_a=*/false, /*reuse_b=*/false);
  }

  float s = c[0] + c[1] + c[2] + c[3] + c[4] + c[5] + c[6] + c[7];
  const float hi = __shfl(s, 16, 32);   // rows 8-15 sit in lane 16
  if (lane == 0) out[0] = (s + hi) * (1.0f / (float)NROWS);
}

// ================= launch =================
extern "C" void kernel_launch(void* const* d_in, const int* in_sizes, int n_in,
                              void* d_out, int out_size, void* d_ws, size_t ws_size,
                              hipStream_t stream) {
  (void)in_sizes; (void)n_in; (void)out_size; (void)ws_size;
  const float* x   = (const float*)d_in[0];
  const int*   tgt = (const int*)d_in[1];
  const float* wm  = (const float*)d_in[2];
  float* partials  = (float*)d_ws;          // NBLK floats = 8 KB scratch

  season_loss_main <<<NBLK, TPB, 0, stream>>>(x, tgt, wm, partials);
  season_loss_final<<<1, 32, 0, stream>>>(partials, (float*)d_out);
}